// FullTensorProductSparse_21303037788628
// MI455X (gfx1250) — compile-verified
//
#include <hip/hip_runtime.h>
#include <cstdint>

// ============================================================================
// Compile-time CG coefficient generation (port of the reference's
// _su2_cg / _q / _wigner_3j, evaluated entirely in constexpr).
// ============================================================================
constexpr double FACT[10] = {1.0, 1.0, 2.0, 6.0, 24.0, 120.0,
                             720.0, 5040.0, 40320.0, 362880.0};
constexpr double cfact(int n) { return FACT[n]; }

constexpr double csqrt(double x) {
  if (x <= 0.0) return 0.0;
  double g = x > 1.0 ? x : 1.0;
  for (int i = 0; i < 64; ++i) {
    double n = 0.5 * (g + x / g);
    if (n == g) break;
    g = n;
  }
  return g;
}
constexpr int imax2(int a, int b) { return a > b ? a : b; }
constexpr int imin2(int a, int b) { return a < b ? a : b; }

constexpr double su2_cg(int j1, int m1, int j2, int m2, int j3, int m3) {
  if (m3 != m1 + m2) return 0.0;
  const int vmin = imax2(imax2(-j1 + j2 + m3, -j1 + m1), 0);
  const int vmax = imin2(imin2(j2 + j3 + m1, j3 - j1 + j2), j3 + m3);
  if (vmax < vmin) return 0.0;
  const double C = csqrt((2.0 * j3 + 1.0) * cfact(j3 + j1 - j2) * cfact(j3 - j1 + j2) *
                         cfact(j1 + j2 - j3) * cfact(j3 + m3) * cfact(j3 - m3) /
                         (cfact(j1 + j2 + j3 + 1) * cfact(j1 - m1) * cfact(j1 + m1) *
                          cfact(j2 - m2) * cfact(j2 + m2)));
  double S = 0.0;
  for (int v = vmin; v <= vmax; ++v) {
    const double sgn = ((v + j2 + m2) & 1) ? -1.0 : 1.0;
    S += sgn / cfact(v) * cfact(j2 + j3 + m1 - v) * cfact(j1 - m1 + v) /
         (cfact(j3 - j1 + j2 - v) * cfact(j3 + m3 - v) * cfact(v + j1 - j2 - m3));
  }
  return C * S;
}

struct Cx { double re; double im; };
constexpr Cx cmul(Cx a, Cx b) { return Cx{a.re * b.re - a.im * b.im, a.re * b.im + a.im * b.re}; }
constexpr Cx cconj(Cx a) { return Cx{a.re, -a.im}; }

struct QM { Cx q[9][9]; };
constexpr QM make_q(int l) {
  QM Q{};
  const double s = 1.0 / csqrt(2.0);
  for (int m = -l; m < 0; ++m) {
    Q.q[l + m][l - m] = Cx{s, 0.0};       // q[l+m, l+|m|]
    Q.q[l + m][l + m] = Cx{0.0, -s};      // q[l+m, l-|m|] = -i*s
  }
  Q.q[l][l] = Cx{1.0, 0.0};
  for (int m = 1; m <= l; ++m) {
    const double f = (m & 1) ? -s : s;    // (-1)^m * s
    Q.q[l + m][l + m] = Cx{f, 0.0};
    Q.q[l + m][l - m] = Cx{0.0, f};       // i*(-1)^m*s
  }
  // multiply by (-i)^l
  const Cx ph = (l % 4 == 0) ? Cx{1.0, 0.0} : (l % 4 == 1) ? Cx{0.0, -1.0}
              : (l % 4 == 2) ? Cx{-1.0, 0.0} : Cx{0.0, 1.0};
  for (int i = 0; i < 2 * l + 1; ++i)
    for (int j = 0; j < 2 * l + 1; ++j) Q.q[i][j] = cmul(ph, Q.q[i][j]);
  return Q;
}

template <int L1, int L2, int L3>
struct CGArr { float v[2 * L1 + 1][2 * L2 + 1][2 * L3 + 1]; };

template <int L1, int L2, int L3>
constexpr CGArr<L1, L2, L3> make_cg() {
  constexpr int d1 = 2 * L1 + 1, d2 = 2 * L2 + 1, d3 = 2 * L3 + 1;
  const QM q1 = make_q(L1), q2 = make_q(L2), q3 = make_q(L3);
  double cgv[d1][d2]{};  // su2 CG hoisted out of the triple loop (m3 = m1+m2 fixed)
  for (int i = 0; i < d1; ++i)
    for (int k = 0; k < d2; ++k) {
      const int mi = (i - L1) + (k - L2) + L3;
      cgv[i][k] = (mi >= 0 && mi < d3) ? su2_cg(L1, i - L1, L2, k - L2, L3, mi - L3) : 0.0;
    }
  double re[d1][d2][d3]{};
  double nrm = 0.0;
  for (int j = 0; j < d1; ++j)
    for (int l = 0; l < d2; ++l)
      for (int n = 0; n < d3; ++n) {
        double sre = 0.0;
        for (int i = 0; i < d1; ++i)
          for (int k = 0; k < d2; ++k) {
            const double cg = cgv[i][k];
            if (cg == 0.0) continue;
            const int mi = (i - L1) + (k - L2) + L3;
            Cx t = cmul(q1.q[i][j], q2.q[k][l]);
            t = cmul(t, cconj(q3.q[mi][n]));
            sre += t.re * cg;
          }
        re[j][l][n] = sre;
        nrm += sre * sre;
      }
  const double scale = csqrt((double)(2 * L3 + 1)) / csqrt(nrm);
  CGArr<L1, L2, L3> r{};
  for (int j = 0; j < d1; ++j)
    for (int l = 0; l < d2; ++l)
      for (int n = 0; n < d3; ++n) r.v[j][l][n] = (float)(re[j][l][n] * scale);
  return r;
}

static_assert(make_cg<0, 0, 0>().v[0][0][0] == 1.0f, "scalar path must be identity");

// ============================================================================
// Kernel: one wave32 per row (lane == mul index u). Results staged in LDS in
// final layout, then streamed out coalesced via CDNA5 async LDS->global b128
// stores with a non-temporal hint (fallback: plain float4 stores).
// ============================================================================
#define NROWS   4      // rows per block (one per wave)
#define THREADS 128
#define ROW_DIM 2592

typedef int v4i __attribute__((ext_vector_type(4)));
#if defined(__gfx1250__)
typedef __attribute__((address_space(1))) v4i* gv4i_ptr;  // global 128-bit
typedef __attribute__((address_space(3))) v4i* lv4i_ptr;  // LDS 128-bit
#endif

// gfx12 CPol immediate: TH in bits [2:0], SCOPE in bits [4:3]. TH_NT = 1.
// Output is a 1 GB streaming write that is never re-read: keep it out of L2.
#define CPOL_STORE_NT 1

__device__ __forceinline__ void store_b128_from_lds(float* g, const float* l) {
#if defined(__gfx1250__) && __has_builtin(__builtin_amdgcn_global_store_async_from_lds_b128)
  __builtin_amdgcn_global_store_async_from_lds_b128(
      (gv4i_ptr)(void*)g, (lv4i_ptr)(void*)(float*)l, 0, CPOL_STORE_NT);
#else
  *(float4*)g = *(const float4*)l;
#endif
}

__device__ __forceinline__ void wait_async_done() {
#if defined(__gfx1250__) && __has_builtin(__builtin_amdgcn_global_store_async_from_lds_b128)
#if __has_builtin(__builtin_amdgcn_s_wait_asynccnt)
  __builtin_amdgcn_s_wait_asynccnt(0);
#else
  asm volatile("s_wait_asynccnt 0" ::: "memory");
#endif
#endif
}

template <int L1, int L2, int L3>
__device__ __forceinline__ void tp_path(const float* __restrict__ x1,  // &x1[Mbase]
                                        const float* __restrict__ x2,  // &x2[nbase]
                                        float* __restrict__ sm,        // wave LDS base
                                        int outOff, int lane) {
  constexpr int d1 = 2 * L1 + 1, d2 = 2 * L2 + 1, d3 = 2 * L3 + 1;
  constexpr CGArr<L1, L2, L3> cg = make_cg<L1, L2, L3>();
  float acc[d3];
#pragma unroll
  for (int o = 0; o < d3; ++o) acc[o] = 0.0f;
#pragma unroll
  for (int m = 0; m < d1; ++m) {
#pragma unroll
    for (int n = 0; n < d2; ++n) {
      const float p = x1[m] * x2[n];
#pragma unroll
      for (int o = 0; o < d3; ++o) {
        const float c = cg.v[m][n][o];
        if (c != 0.0f) acc[o] = fmaf(c, p, acc[o]);  // folds away for zero CG
      }
    }
  }
  // output chunk layout: element (u, o) -> u*d3 + o  (u-major), odd stride -> no LDS bank conflicts
#pragma unroll
  for (int o = 0; o < d3; ++o) sm[outOff + lane * d3 + o] = acc[o];
}

__global__ void __launch_bounds__(THREADS)
ftp_kernel(const float* __restrict__ in1, const float* __restrict__ in2,
           float* __restrict__ out, int N) {
  __shared__ __align__(16) float sm[NROWS * ROW_DIM];  // 41.5 KB of the 320 KB WGP LDS
  const int tid  = threadIdx.x;
  const int lane = tid & 31;
  const int wave = tid >> 5;
  const int row0 = blockIdx.x * NROWS;
  const int row  = row0 + wave;

  if (row < N) {
    const float* r1 = in1 + (size_t)row * 288;
    const float* r2 = in2 + (size_t)row * 9;
    float x1[9], x2[9];
#pragma unroll
    for (int m = 0; m < 9; ++m) x1[m] = r1[m * 32 + lane];  // coalesced 128B loads
#pragma unroll
    for (int n = 0; n < 9; ++n) x2[n] = r2[n];              // uniform broadcast loads

    float* w = sm + wave * ROW_DIM;
    // e3nn sorted output order: key (l3, -p*(-1)^l3, path index)
    tp_path<0, 0, 0>(&x1[0], &x2[0], w,    0, lane);
    tp_path<1, 1, 0>(&x1[1], &x2[1], w,   32, lane);
    tp_path<2, 2, 0>(&x1[4], &x2[4], w,   64, lane);
    tp_path<0, 1, 1>(&x1[0], &x2[1], w,   96, lane);
    tp_path<1, 0, 1>(&x1[1], &x2[0], w,  192, lane);
    tp_path<1, 2, 1>(&x1[1], &x2[4], w,  288, lane);
    tp_path<2, 1, 1>(&x1[4], &x2[1], w,  384, lane);
    tp_path<1, 1, 1>(&x1[1], &x2[1], w,  480, lane);
    tp_path<2, 2, 1>(&x1[4], &x2[4], w,  576, lane);
    tp_path<0, 2, 2>(&x1[0], &x2[4], w,  672, lane);
    tp_path<1, 1, 2>(&x1[1], &x2[1], w,  832, lane);
    tp_path<2, 0, 2>(&x1[4], &x2[0], w,  992, lane);
    tp_path<2, 2, 2>(&x1[4], &x2[4], w, 1152, lane);
    tp_path<1, 2, 2>(&x1[1], &x2[4], w, 1312, lane);
    tp_path<2, 1, 2>(&x1[4], &x2[1], w, 1472, lane);
    tp_path<1, 2, 3>(&x1[1], &x2[4], w, 1632, lane);
    tp_path<2, 1, 3>(&x1[4], &x2[1], w, 1856, lane);
    tp_path<2, 2, 3>(&x1[4], &x2[4], w, 2080, lane);
    tp_path<2, 2, 4>(&x1[4], &x2[4], w, 2304, lane);
  }
  __syncthreads();

  // Stream NROWS contiguous rows (NROWS*2592 floats) to global, 16B per lane.
  int nrows = N - row0;
  if (nrows > 0) {
    if (nrows > NROWS) nrows = NROWS;
    const int nf4 = nrows * (ROW_DIM / 4);
    float* gbase = out + (size_t)row0 * ROW_DIM;
    for (int f = tid; f < nf4; f += THREADS) {
      store_b128_from_lds(gbase + f * 4, sm + f * 4);
    }
  }
  wait_async_done();
}

extern "C" void kernel_launch(void* const* d_in, const int* in_sizes, int n_in,
                              void* d_out, int out_size, void* d_ws, size_t ws_size,
                              hipStream_t stream) {
  const float* in1 = (const float*)d_in[0];
  const float* in2 = (const float*)d_in[1];
  float* out = (float*)d_out;
  const int N = in_sizes[0] / 288;
  const int blocks = (N + NROWS - 1) / NROWS;
  ftp_kernel<<<blocks, THREADS, 0, stream>>>(in1, in2, out, N);
}